// HybridIPGLayer_67164698575285
// MI455X (gfx1250) — compile-verified
//
#include <hip/hip_runtime.h>
#include <math.h>

typedef float v2f __attribute__((ext_vector_type(2)));
typedef float v8f __attribute__((ext_vector_type(8)));

#define BB 2
#define CC 64
#define HH 256
#define WW 256
#define HW (HH*WW)
#define KEXP 16

static __device__ __forceinline__ v8f wmma4(v2f a, v2f b, v8f c) {
  // D = A(16x4 f32) * B(4x16 f32) + C(16x16 f32)
  return __builtin_amdgcn_wmma_f32_16x16x4_f32(false, a, false, b, (short)0, c, false, false);
}

// ---------------- stats init ----------------
__global__ void k_init(unsigned* st) {
  if (threadIdx.x < 2) st[threadIdx.x] = 0x7f7fffffu;       // dfmin = +FLT_MAX
  else if (threadIdx.x < 4) st[threadIdx.x] = 0u;           // dfmax = 0
}

// ---------------- weight transpose: [k][co][ci][3][3] -> [k][tap][ci][co] ----------------
__global__ __launch_bounds__(256) void k_wT(const float* __restrict__ w, float* __restrict__ wT) {
  int i = blockIdx.x * 256 + threadIdx.x;           // dst index
  int co = i & 63;
  int ci = (i >> 6) & 63;
  int kt = i >> 12;                                  // k*9 + t
  int t  = kt % 9;
  int k  = kt / 9;
  wT[i] = w[(((size_t)(k * 64 + co)) * 64 + ci) * 9 + t];
}

// ---------------- 2x2 avg pool ----------------
__global__ __launch_bounds__(256) void k_down(const float* __restrict__ x, float* __restrict__ xd) {
  int i = blockIdx.x * 256 + threadIdx.x;            // over B*C*128*128
  int j = i & (128 * 128 - 1);
  int bc = i >> 14;
  int r = j >> 7, c2 = j & 127;
  const float* p = x + ((size_t)bc * HH + 2 * r) * WW + 2 * c2;
  xd[i] = 0.25f * (p[0] + p[1] + p[WW] + p[WW + 1]);
}

// ---------------- df (L1 diff vs bilinear up) + channel L2 inv-norm + df min/max ----------------
__global__ __launch_bounds__(256) void k_dfinv(const float* __restrict__ x, const float* __restrict__ xd,
                                               float* __restrict__ inv, float* __restrict__ df,
                                               unsigned* __restrict__ st) {
  int pix = blockIdx.x * 256 + threadIdx.x;          // B*H*W
  int b = pix / HW;
  int hw = pix - b * HW;
  int h = hw / WW, w = hw - (hw / WW) * WW;
  // half-pixel bilinear (align_corners=False), scale 2
  int iy0 = (h >> 1) + ((h & 1) ? 0 : -1); float ty = (h & 1) ? 0.25f : 0.75f;
  int ix0 = (w >> 1) + ((w & 1) ? 0 : -1); float tx = (w & 1) ? 0.25f : 0.75f;
  int iy0c = iy0 < 0 ? 0 : iy0;       int iy1c = (iy0 + 1) > 127 ? 127 : (iy0 + 1);
  int ix0c = ix0 < 0 ? 0 : ix0;       int ix1c = (ix0 + 1) > 127 ? 127 : (ix0 + 1);
  float w00 = (1.f - ty) * (1.f - tx), w01 = (1.f - ty) * tx;
  float w10 = ty * (1.f - tx),         w11 = ty * tx;
  float s = 0.f, ss = 0.f;
  for (int c = 0; c < CC; ++c) {
    float xv = x[((size_t)(b * CC + c) * HH + h) * WW + w];
    const float* xp = xd + (size_t)(b * CC + c) * 128 * 128;
    float u = w00 * xp[iy0c * 128 + ix0c] + w01 * xp[iy0c * 128 + ix1c]
            + w10 * xp[iy1c * 128 + ix0c] + w11 * xp[iy1c * 128 + ix1c];
    s += fabsf(xv - u);
    ss += xv * xv;
  }
  df[pix] = s;
  inv[pix] = 1.f / fmaxf(sqrtf(ss), 1e-12f);
  __shared__ float rmin[256], rmax[256];
  rmin[threadIdx.x] = s; rmax[threadIdx.x] = s;
  __syncthreads();
  for (int o = 128; o > 0; o >>= 1) {
    if (threadIdx.x < o) {
      rmin[threadIdx.x] = fminf(rmin[threadIdx.x], rmin[threadIdx.x + o]);
      rmax[threadIdx.x] = fmaxf(rmax[threadIdx.x], rmax[threadIdx.x + o]);
    }
    __syncthreads();
  }
  if (threadIdx.x == 0) {   // df >= 0 so uint bit order == float order
    atomicMin(&st[b],     __float_as_uint(rmin[0]));
    atomicMax(&st[2 + b], __float_as_uint(rmax[0]));
  }
}

// ---------------- GroupNorm stats: one block per (b, group) ----------------
__global__ __launch_bounds__(256) void k_gnstats(const float* __restrict__ x, float* __restrict__ gnstat) {
  int b = blockIdx.x >> 5, g = blockIdx.x & 31;
  const float* p = x + (size_t)(b * CC + 2 * g) * HW;   // 2 contiguous channels
  float s = 0.f, ss = 0.f;
  for (int i = threadIdx.x; i < 2 * HW; i += 256) { float v = p[i]; s += v; ss += v * v; }
  __shared__ float r1[256], r2[256];
  r1[threadIdx.x] = s; r2[threadIdx.x] = ss;
  __syncthreads();
  for (int o = 128; o > 0; o >>= 1) {
    if (threadIdx.x < o) { r1[threadIdx.x] += r1[threadIdx.x + o]; r2[threadIdx.x] += r2[threadIdx.x + o]; }
    __syncthreads();
  }
  if (threadIdx.x == 0) {
    float n = 2.f * HW;
    float mu = r1[0] / n;
    float var = r2[0] / n - mu * mu;
    gnstat[b * 32 + g] = mu;
    gnstat[64 + b * 32 + g] = rsqrtf(var + 1e-5f);
  }
}

// ---------------- per-pixel expert index ----------------
__global__ __launch_bounds__(256) void k_counts(const float* __restrict__ df, const unsigned* __restrict__ st,
                                                int* __restrict__ kidx) {
  int pix = blockIdx.x * 256 + threadIdx.x;
  int b = pix / HW;
  float dmin = __uint_as_float(st[b]);
  float dmax = __uint_as_float(st[2 + b]);
  float dp = (df[pix] - dmin) / (dmax - dmin + 1e-8f);
  dp = dp * dp;
  int cam = (int)rintf(dp * 15.f);          // RNE matches jnp.round
  if (cam < 0) cam = 0;
  kidx[pix] = (dp > 0.9f) ? cam : 0;        // == counts - 1
}

// ---------------- expert conv (WMMA f32) + GroupNorm apply -> enhanced ----------------
// Block tile: 64 pixels of one row. 8 waves = 4 co-tiles x 2 segment-pairs.
// Each wave: 16 out-channels x 32 pixels (two 16-pixel WMMA N-tiles sharing A registers).
__global__ __launch_bounds__(256) void k_conv(const float* __restrict__ x, const float* __restrict__ wT,
                                              const float* __restrict__ inv, const int* __restrict__ kidx,
                                              const float* __restrict__ gn_w, const float* __restrict__ gn_b,
                                              const float* __restrict__ gnstat, float* __restrict__ enh) {
  __shared__ float XT[64][3][68];   // pre-normalized x tile + halo (zero-padded), 66 cols used
  __shared__ float IV[3][68];       // inv-norm tile (zero outside)
  __shared__ int   KS[64];

  const int wtiles = WW / 64;       // 4
  int blk = blockIdx.x;
  int b   = blk / (HH * wtiles);
  int rem = blk - b * (HH * wtiles);
  int h   = rem / wtiles;
  int w0  = (rem - h * wtiles) * 64;
  int tid = threadIdx.x;

  for (int i = tid; i < 3 * 66; i += 256) {
    int r = i / 66, c2 = i - (i / 66) * 66;
    int hh = h - 1 + r, ww = w0 - 1 + c2;
    float v = 0.f;
    if (hh >= 0 && hh < HH && ww >= 0 && ww < WW) v = inv[(b * HH + hh) * WW + ww];
    IV[r][c2] = v;
  }
  if (tid < 64) KS[tid] = kidx[(b * HH + h) * WW + w0 + tid];
  __syncthreads();                                   // IV ready before XT uses it
  for (int i = tid; i < 64 * 3 * 66; i += 256) {
    int ci = i / 198;
    int rc = i - ci * 198;
    int r = rc / 66, c2 = rc - (rc / 66) * 66;
    int hh = h - 1 + r, ww = w0 - 1 + c2;
    float v = 0.f;
    if (hh >= 0 && hh < HH && ww >= 0 && ww < WW) v = x[((size_t)(b * CC + ci) * HH + hh) * WW + ww];
    XT[ci][r][c2] = v * IV[r][c2];                   // stage xn = x * invnorm
  }
  __syncthreads();

  int wv   = tid >> 5;
  int lane = tid & 31;
  int cot  = wv & 3;         // out-channel tile (16 ch)
  int sp   = wv >> 2;        // segment pair: pixels 32*sp .. 32*sp+31
  int p    = lane & 15;      // serves as A's M and B's N
  int hi   = (lane < 16) ? 0 : 2;

  unsigned km = 0;
  for (int i = 0; i < 32; ++i) km |= (1u << KS[sp * 32 + i]);   // wave-uniform
  int myk0 = KS[sp * 32 + p];
  int myk1 = KS[sp * 32 + 16 + p];

  v8f acc0 = {0.f, 0.f, 0.f, 0.f, 0.f, 0.f, 0.f, 0.f};
  v8f acc1 = {0.f, 0.f, 0.f, 0.f, 0.f, 0.f, 0.f, 0.f};
  unsigned m = km;
  while (m) {
    int k = __ffs(m) - 1;
    m &= m - 1;
    // warm L2 for this expert's weight block (global_prefetch path)
    __builtin_prefetch(wT + (size_t)k * 9 * 4096 + (tid << 6), 0, 1);
    v8f d0 = {0.f, 0.f, 0.f, 0.f, 0.f, 0.f, 0.f, 0.f};
    v8f d1 = {0.f, 0.f, 0.f, 0.f, 0.f, 0.f, 0.f, 0.f};
#pragma unroll 1
    for (int t = 0; t < 9; ++t) {                    // keep one tap's operands live
      int dy = t / 3, dx = t - dy * 3;
      const float* wb = wT + (size_t)(k * 9 + t) * 4096 + (cot * 16 + p);  // + ci*64
      int col0 = sp * 32 + p + dx;                   // [0..65]
      int col1 = col0 + 16;
      v2f a[16];
#pragma unroll
      for (int kb = 0; kb < 16; ++kb) {              // batched A loads (one clause)
        int ci = kb * 4 + hi;
        a[kb].x = wb[ci * 64];
        a[kb].y = wb[(ci + 1) * 64];
      }
#pragma unroll
      for (int kb = 0; kb < 16; ++kb) {              // each A pair feeds two WMMAs
        int ci = kb * 4 + hi;
        v2f b0, b1;
        b0.x = XT[ci][dy][col0];
        b0.y = XT[ci + 1][dy][col0];
        b1.x = XT[ci][dy][col1];
        b1.y = XT[ci + 1][dy][col1];
        d0 = wmma4(a[kb], b0, d0);
        d1 = wmma4(a[kb], b1, d1);
      }
    }
    bool s0 = (myk0 == k), s1 = (myk1 == k);
#pragma unroll
    for (int i = 0; i < 8; ++i) {
      acc0[i] = s0 ? d0[i] : acc0[i];
      acc1[i] = s1 ? d1[i] : acc1[i];
    }
  }

  // epilogue: + GroupNorm(identity); raw x reloaded from global (coalesced)
  int wpix0 = w0 + sp * 32 + p;
  int wpix1 = wpix0 + 16;
#pragma unroll
  for (int i = 0; i < 8; ++i) {
    int c = cot * 16 + i + ((lane < 16) ? 0 : 8);     // D: M = i + (lane<16?0:8)
    int g = c >> 1;
    float mu = gnstat[b * 32 + g];
    float rs = gnstat[64 + b * 32 + g];
    float a0 = rs * gn_w[c];
    size_t base = (size_t)(b * CC + c) * HW + h * WW;
    float xv0 = x[base + wpix0];
    float xv1 = x[base + wpix1];
    enh[base + wpix0] = acc0[i] + (xv0 - mu) * a0 + gn_b[c];
    enh[base + wpix1] = acc1[i] + (xv1 - mu) * a0 + gn_b[c];
  }
}

// ---------------- FFN (WMMA f32): out = enh + W2*relu(W1*enh + b1) + b2 ----------------
__global__ __launch_bounds__(256) void k_ffn(const float* __restrict__ enh,
                                             const float* __restrict__ w1, const float* __restrict__ b1,
                                             const float* __restrict__ w2, const float* __restrict__ b2,
                                             float* __restrict__ out) {
  __shared__ float W1L[128][64];
  __shared__ float W2L[64][128];
  __shared__ float B1L[128];
  __shared__ float B2L[64];
  __shared__ float HL[8][128][16];

  int tid = threadIdx.x;
  for (int i = tid; i < 128 * 64; i += 256) W1L[i >> 6][i & 63] = w1[i];
  for (int i = tid; i < 64 * 128; i += 256) W2L[i >> 7][i & 127] = w2[i];
  if (tid < 128) B1L[tid] = b1[tid];
  if (tid < 64)  B2L[tid] = b2[tid];
  __syncthreads();

  int wv = tid >> 5, lane = tid & 31;
  int p  = lane & 15;
  int hi = (lane < 16) ? 0 : 2;
  long pix0 = (long)blockIdx.x * 128 + wv * 16;       // 16 consecutive pixels (one row)
  long b  = pix0 / HW;
  long hw = pix0 - b * HW;
  const float* eb = enh + (size_t)b * CC * HW + hw;   // + c*HW + p

  // GEMM1: h(128x16) = W1(128x64) * E(64x16)
  v8f h8[8];
#pragma unroll
  for (int co = 0; co < 8; ++co) h8[co] = (v8f){0.f, 0.f, 0.f, 0.f, 0.f, 0.f, 0.f, 0.f};
  for (int kb = 0; kb < 16; ++kb) {
    int ci = kb * 4 + hi;
    v2f bb;
    bb.x = eb[(size_t)ci * HW + p];
    bb.y = eb[(size_t)(ci + 1) * HW + p];
#pragma unroll
    for (int co = 0; co < 8; ++co) {
      v2f a;
      a.x = W1L[co * 16 + p][ci];
      a.y = W1L[co * 16 + p][ci + 1];
      h8[co] = wmma4(a, bb, h8[co]);
    }
  }
  // bias + ReLU -> LDS (wave-local)
#pragma unroll
  for (int co = 0; co < 8; ++co)
#pragma unroll
    for (int i = 0; i < 8; ++i) {
      int c = co * 16 + i + ((lane < 16) ? 0 : 8);
      HL[wv][c][p] = fmaxf(h8[co][i] + B1L[c], 0.f);
    }

  // GEMM2: o(64x16) = W2(64x128) * h(128x16), + b2 + residual
#pragma unroll
  for (int co = 0; co < 4; ++co) {
    v8f dacc = {0.f, 0.f, 0.f, 0.f, 0.f, 0.f, 0.f, 0.f};
    for (int kb = 0; kb < 32; ++kb) {
      int ci = kb * 4 + hi;
      v2f a, bb;
      a.x = W2L[co * 16 + p][ci];
      a.y = W2L[co * 16 + p][ci + 1];
      bb.x = HL[wv][ci][p];
      bb.y = HL[wv][ci + 1][p];
      dacc = wmma4(a, bb, dacc);
    }
#pragma unroll
    for (int i = 0; i < 8; ++i) {
      int c = co * 16 + i + ((lane < 16) ? 0 : 8);
      float e = eb[(size_t)c * HW + p];
      out[((size_t)b * CC + c) * HW + hw + p] = e + dacc[i] + B2L[c];
    }
  }
}

extern "C" void kernel_launch(void* const* d_in, const int* in_sizes, int n_in,
                              void* d_out, int out_size, void* d_ws, size_t ws_size,
                              hipStream_t stream) {
  const float* x    = (const float*)d_in[0];
  const float* kw   = (const float*)d_in[1];
  const float* gn_w = (const float*)d_in[2];
  const float* gn_b = (const float*)d_in[3];
  const float* w1   = (const float*)d_in[4];
  const float* b1   = (const float*)d_in[5];
  const float* w2   = (const float*)d_in[6];
  const float* b2   = (const float*)d_in[7];
  float* out = (float*)d_out;
  float* ws  = (float*)d_ws;

  // workspace layout (floats)
  size_t o_xd  = 0;                       // B*C*128*128
  size_t o_inv = o_xd + (size_t)BB * CC * 128 * 128;
  size_t o_df  = o_inv + (size_t)BB * HW;
  size_t o_kid = o_df + (size_t)BB * HW;
  size_t o_enh = o_kid + (size_t)BB * HW;
  size_t o_wT  = o_enh + (size_t)BB * CC * HW;
  size_t o_st  = o_wT + (size_t)KEXP * 9 * 64 * 64;

  float*    xd   = ws + o_xd;
  float*    inv  = ws + o_inv;
  float*    df   = ws + o_df;
  int*      kid  = (int*)(ws + o_kid);
  float*    enh  = ws + o_enh;
  float*    wT   = ws + o_wT;
  unsigned* st   = (unsigned*)(ws + o_st);    // [0..1] dfmin bits, [2..3] dfmax bits
  float*    gst  = ws + o_st + 4;             // [0..63] mu, [64..127] rstd

  k_init<<<1, 64, 0, stream>>>(st);
  k_wT<<<(KEXP * 9 * 64 * 64) / 256, 256, 0, stream>>>(kw, wT);
  k_down<<<(BB * CC * 128 * 128) / 256, 256, 0, stream>>>(x, xd);
  k_dfinv<<<(BB * HW) / 256, 256, 0, stream>>>(x, xd, inv, df, st);
  k_gnstats<<<BB * 32, 256, 0, stream>>>(x, gst);
  k_counts<<<(BB * HW) / 256, 256, 0, stream>>>(df, st, kid);
  k_conv<<<BB * HH * (WW / 64), 256, 0, stream>>>(x, wT, inv, kid, gn_w, gn_b, gst, enh);
  k_ffn<<<(BB * HW) / 128, 256, 0, stream>>>(enh, w1, b1, w2, b2, out);
}